// InceptionPlus_17093969838162
// MI455X (gfx1250) — compile-verified
//
#include <hip/hip_runtime.h>
#include <hip/hip_bf16.h>

#define B_ROWS       128
#define C_COLS       65536
#define KTOP         200
#define MW           16
#define KM           (KTOP * MW)      // 3200
#define H_DIM        512
#define NUM_LABELS_P 670000
#define CBUF         512              // candidate collection buffer (>= KTOP + tie slack)

typedef __attribute__((ext_vector_type(2))) float v2f;
typedef __attribute__((ext_vector_type(8))) float v8f;

__device__ __forceinline__ unsigned int sortable_key(float x) {
    unsigned int u = __float_as_uint(x);
    return (u & 0x80000000u) ? ~u : (u | 0x80000000u);   // descending order == larger key
}
__device__ __forceinline__ float key_to_float(unsigned int k) {
    unsigned int u = (k & 0x80000000u) ? (k & 0x7FFFFFFFu) : ~k;
    return __uint_as_float(u);
}
__device__ __forceinline__ float sigmoidf(float x) {
    return 1.0f / (1.0f + __expf(-x));
}

// ---------------------------------------------------------------------------
// Kernel 1: exact per-row top-200 via 4-pass byte radix-select.
// One 256-thread workgroup per batch row.
// ---------------------------------------------------------------------------
__global__ __launch_bounds__(256) void topk_kernel(
    const float* __restrict__ meta,          // [128, 65536]
    float* __restrict__ tvals,               // [128, 200] sigmoid(logit), sorted desc
    int*   __restrict__ tidx)                // [128, 200] column indices
{
    const int b = blockIdx.x;
    const int t = threadIdx.x;
    const float* row = meta + (size_t)b * C_COLS;

    __shared__ unsigned int hist[256];
    __shared__ unsigned int s_prefix, s_mask;
    __shared__ int s_remaining, s_count;
    __shared__ unsigned int ckey[CBUF];
    __shared__ int cidx[CBUF];

    // default-fill outputs (pathological tie-overflow safety)
    for (int j = t; j < KTOP; j += 256) {
        tvals[b * KTOP + j] = 0.0f;
        tidx[b * KTOP + j]  = 0;
    }
    if (t == 0) { s_prefix = 0u; s_mask = 0u; s_remaining = KTOP; }
    __syncthreads();

    // 4 radix passes, high byte to low byte
    for (int level = 3; level >= 0; --level) {
        hist[t] = 0u;
        __syncthreads();
        const unsigned int pref = s_prefix, msk = s_mask;
        const int shift = level * 8;
        for (int i = t; i < C_COLS; i += 256) {
            unsigned int k = sortable_key(row[i]);
            if ((k & msk) == pref)
                atomicAdd(&hist[(k >> shift) & 0xFFu], 1u);
        }
        __syncthreads();
        if (t == 0) {
            int rem = s_remaining, cum = 0, sel = 0;
            for (int bin = 255; bin >= 0; --bin) {
                int c2 = cum + (int)hist[bin];
                if (c2 >= rem) { sel = bin; break; }
                cum = c2;
            }
            s_remaining = rem - cum;                       // still needed at == sel
            s_prefix    = pref | ((unsigned int)sel << shift);
            s_mask      = msk | (0xFFu << shift);
        }
        __syncthreads();
    }
    const unsigned int thr = s_prefix;                     // exact k-th largest key
    if (t == 0) s_count = 0;
    __syncthreads();

    // collect all keys >= threshold (count >= KTOP by construction)
    for (int i = t; i < C_COLS; i += 256) {
        unsigned int k = sortable_key(row[i]);
        if (k >= thr) {
            int p = atomicAdd(&s_count, 1);
            if (p < CBUF) { ckey[p] = k; cidx[p] = i; }
        }
    }
    __syncthreads();
    int n = s_count; if (n > CBUF) n = CBUF;

    // rank each collected element: desc by key, ties -> lower index first
    for (int j = t; j < n; j += 256) {
        const unsigned int kj = ckey[j];
        const int ij = cidx[j];
        int rank = 0;
        for (int i = 0; i < n; ++i) {
            unsigned int ki = ckey[i];
            if (ki > kj || (ki == kj && cidx[i] < ij)) rank++;
        }
        if (rank < KTOP) {
            tvals[b * KTOP + rank] = sigmoidf(key_to_float(kj));
            tidx[b * KTOP + rank]  = ij;
        }
    }
}

// ---------------------------------------------------------------------------
// Kernel 2: cluster gather + stable compaction of padding to the end.
// One 256-thread workgroup per batch row.
// ---------------------------------------------------------------------------
__global__ __launch_bounds__(256) void gather_compact_kernel(
    const float* __restrict__ tvals,         // [128, 200]
    const int*   __restrict__ tidx,          // [128, 200]
    const int*   __restrict__ clusters,      // [65536, 16] int32 view
    int*   __restrict__ cands,               // [128, 3200] ws
    float* __restrict__ cscores,             // [128, 3200] ws
    float* __restrict__ out_cands)           // [128, 3200] output segment 0 (as float)
{
    const int b = blockIdx.x;
    const int t = threadIdx.x;

    __shared__ int   s_cand[KM];
    __shared__ float s_score[KM];
    __shared__ int   s_scan[256];

    for (int p = t; p < KM; p += 256) {
        int j = p >> 4;                      // top-k slot
        int m = p & 15;                      // within-cluster
        int lab = tidx[b * KTOP + j];
        if (lab < 0) lab = 0;
        if (lab >= C_COLS) lab = C_COLS - 1;
        s_cand[p]  = clusters[(size_t)lab * MW + m];
        s_score[p] = tvals[b * KTOP + j];
    }
    __syncthreads();

    // per-thread chunk [p0, p1)
    const int CHUNK = 13;                    // 256*13 = 3328 >= 3200
    int p0 = t * CHUNK; if (p0 > KM) p0 = KM;
    int p1 = p0 + CHUNK; if (p1 > KM) p1 = KM;

    int localValid = 0;
    for (int p = p0; p < p1; ++p)
        localValid += (s_cand[p] != NUM_LABELS_P) ? 1 : 0;
    s_scan[t] = localValid;
    __syncthreads();
    for (int off = 1; off < 256; off <<= 1) {
        int add = (t >= off) ? s_scan[t - off] : 0;
        __syncthreads();
        s_scan[t] += add;
        __syncthreads();
    }
    const int totalValid = s_scan[255];
    const int baseValid  = s_scan[t] - localValid;   // exclusive prefix

    int run = 0;
    for (int p = p0; p < p1; ++p) {
        int  c     = s_cand[p];
        bool valid = (c != NUM_LABELS_P);
        int  vb    = baseValid + run;                 // #valid strictly before p
        int  np    = valid ? vb : (totalValid + (p - vb));
        float sc   = valid ? s_score[p] : 0.0f;
        size_t o = (size_t)b * KM + np;
        cands[o]     = c;
        cscores[o]   = sc;
        out_cands[o] = (float)c;
        if (valid) run++;
    }
}

// ---------------------------------------------------------------------------
// Kernel 3: embedding gather + 16-candidate x 512 dot against h_c[b] using
// V_WMMA_F32_16X16X4_F32 (A = 16x4 f32 tile of embeddings, B = h_c replicated
// across 16 columns, 128 WMMAs over K=512). One wave (32 lanes) per tile.
// ---------------------------------------------------------------------------
__global__ __launch_bounds__(32) void wmma_dot_kernel(
    const float* __restrict__ embed,         // [670001, 512]
    const float* __restrict__ hc,            // [128, 512]
    const int*   __restrict__ cands,         // [128, 3200]
    const float* __restrict__ cscores,       // [128, 3200]
    float* __restrict__ out_scores,          // output segment 1
    float* __restrict__ out_prod)            // output segment 2
{
    const int b    = blockIdx.y;
    const int base = blockIdx.x * MW;        // candidate tile start
    const int lane = threadIdx.x;

    // A layout (32-bit 16x4): lanes 0-15 -> M=lane, K={0,1}; lanes 16-31 -> M=lane-16, K={2,3}
    const int rm   = lane & 15;
    const int koff = (lane >> 4) * 2;

    int c = cands[(size_t)b * KM + base + rm];
    if (c < 0) c = 0;
    if (c > NUM_LABELS_P) c = NUM_LABELS_P;
    const float* arow = embed + (size_t)c * H_DIM;
    const float* hrow = hc + (size_t)b * H_DIM;
    __builtin_prefetch(arow, 0, 1);          // global_prefetch_b8 on the gathered row

    v8f acc = {};
#if __has_builtin(__builtin_amdgcn_wmma_f32_16x16x4_f32)
    #pragma unroll 8
    for (int kb = 0; kb < H_DIM; kb += 4) {
        v2f a  = *(const v2f*)(arow + kb + koff);
        // B (4x16 f32): VGPR0 = rows K=0 (lanes 0-15) / K=2 (lanes 16-31), VGPR1 = K=1 / K=3.
        // h_c replicated across all N columns -> same K offsets as A.
        v2f bb = *(const v2f*)(hrow + kb + koff);
        acc = __builtin_amdgcn_wmma_f32_16x16x4_f32(
            /*neg_a=*/false, a, /*neg_b=*/false, bb,
            /*c_mod=*/(short)0, acc, /*reuse_a=*/false, /*reuse_b=*/false);
    }
    // D (f32 16x16): lane 0 column holds M=0..7 in VGPRs 0..7; lane 16 holds M=8..15.
    if (lane == 0 || lane == 16) {
        const int mbase = (lane == 0) ? 0 : 8;
        #pragma unroll
        for (int r = 0; r < 8; ++r) {
            float logit = acc[r];
            float s = (logit == 0.0f) ? 0.0f : sigmoidf(logit);   // where(==0, -inf) -> 0
            size_t o = (size_t)b * KM + base + mbase + r;
            float csc = cscores[o];
            out_scores[o] = s;
            out_prod[o]   = s * csc;
        }
    }
#else
    // Fallback (should not be taken on gfx1250): plain FMA dot per lane pair.
    float sum = 0.0f;
    for (int kb = 0; kb < H_DIM; kb += 4) {
        sum = fmaf(arow[kb + koff],     hrow[kb + koff],     sum);
        sum = fmaf(arow[kb + koff + 1], hrow[kb + koff + 1], sum);
    }
    // combine the two half-K partial sums (lane L and L+16)
    float other = __shfl_xor(sum, 16, 32);
    float logit = sum + other;
    if (lane < 16) {
        float s = (logit == 0.0f) ? 0.0f : sigmoidf(logit);
        size_t o = (size_t)b * KM + base + lane;
        float csc = cscores[o];
        out_scores[o] = s;
        out_prod[o]   = s * csc;
    }
    (void)acc;
#endif
}

// ---------------------------------------------------------------------------
extern "C" void kernel_launch(void* const* d_in, const int* in_sizes, int n_in,
                              void* d_out, int out_size, void* d_ws, size_t ws_size,
                              hipStream_t stream) {
    const float* meta     = (const float*)d_in[0];   // [128, 65536]
    const float* hc       = (const float*)d_in[1];   // [128, 512]
    const float* embed    = (const float*)d_in[2];   // [670001, 512]
    const int*   clusters = (const int*)d_in[3];     // [65536, 16]
    (void)d_in; (void)in_sizes; (void)n_in; (void)ws_size;
    // topk scalar (d_in[4]) is fixed at 200 per the reference setup.

    // workspace layout
    float* tvals   = (float*)d_ws;                         // 128*200
    int*   tidx    = (int*)(tvals + B_ROWS * KTOP);        // 128*200
    int*   cands   = (int*)(tidx + B_ROWS * KTOP);         // 128*3200
    float* cscores = (float*)(cands + (size_t)B_ROWS * KM);// 128*3200

    // output layout: [cands-as-float | cand_scores | cand_scores*cscores]
    float* out0 = (float*)d_out;
    float* out1 = out0 + (size_t)B_ROWS * KM;
    float* out2 = out1 + (size_t)B_ROWS * KM;
    (void)out_size;

    topk_kernel<<<dim3(B_ROWS), dim3(256), 0, stream>>>(meta, tvals, tidx);
    gather_compact_kernel<<<dim3(B_ROWS), dim3(256), 0, stream>>>(
        tvals, tidx, clusters, cands, cscores, out0);
    wmma_dot_kernel<<<dim3(KM / MW, B_ROWS), dim3(32), 0, stream>>>(
        embed, hc, cands, cscores, out1, out2);
}